// QueryCrossAttentionBase_64991445123362
// MI455X (gfx1250) — compile-verified
//
#include <hip/hip_runtime.h>
#include <math.h>

typedef float v2f __attribute__((ext_vector_type(2)));
typedef float v8f __attribute__((ext_vector_type(8)));

#define N_TOK 16384
#define CDIM  192
#define HEADS 8
#define HD    24
#define NB    8
#define CH    64          // tokens staged per LDS chunk (gram kernel)
#define LPAD  65          // padded LDS row stride (odd -> conflict free)
#define SLICE 1024        // tokens per block
#define NCHUNK (SLICE/CH) // 16

// ---- workspace layout (floats) ----
static constexpr size_t OFF_C   = 0;                                  // [8][6][192][192]
static constexpr size_t OFF_R   = OFF_C   + (size_t)NB*6*CDIM*CDIM;   // [8][4][192]
static constexpr size_t OFF_U   = OFF_R   + (size_t)NB*4*CDIM;        // [8][6][192][192]
static constexpr size_t OFF_A   = OFF_U   + (size_t)NB*6*CDIM*CDIM;   // [8*2][8][24][24]
static constexpr size_t OFF_AWV = OFF_A   + (size_t)NB*2*HEADS*HD*HD; // [16][192][192]
static constexpr size_t OFF_ABV = OFF_AWV + (size_t)NB*2*CDIM*CDIM;   // [16][192]
static constexpr size_t OFF_M   = OFF_ABV + (size_t)NB*2*CDIM;        // [16][192][192]
static constexpr size_t OFF_CV  = OFF_M   + (size_t)NB*2*CDIM*CDIM;   // [16][192]

__device__ __forceinline__ v8f zero8() {
    v8f z; for (int i = 0; i < 8; ++i) z[i] = 0.f; return z;
}

// ============================================================================
// Kernel 1: Gram matrices  C[b][p] = Ya @ Yb^T  over token slices, via WMMA f32
// pairs: 0:(Q1,Q1) 1:(Q2,Q2) 2:(X1,X1) 3:(X2,X2) 4:(Q1,X2) 5:(Q2,X1)
// Each wave owns a 3x6 sub-grid of 16x16 output tiles: per k-step it loads
// 3 A-fragments + 6 B-fragments from LDS and issues 18 WMMAs (0.5 ds/wmma).
// Self-pairs also accumulate row sums r[b][tensor].
// ============================================================================
__global__ __launch_bounds__(256) void gram_kernel(
    const float* __restrict__ x1, const float* __restrict__ x2,
    const float* __restrict__ q1, const float* __restrict__ q2,
    float* __restrict__ Cacc, float* __restrict__ Racc)
{
    __shared__ float As[CDIM * LPAD];
    __shared__ float Bs[CDIM * LPAD];

    const int slice = blockIdx.x;   // 0..15 token slices of 1024
    const int pair  = blockIdx.y;   // 0..5
    const int b     = blockIdx.z;   // 0..7

    const float* ta; const float* tb; int rsel = 0;
    switch (pair) {
        case 0: ta = q1; tb = q1; rsel = 2; break;
        case 1: ta = q2; tb = q2; rsel = 3; break;
        case 2: ta = x1; tb = x1; rsel = 0; break;
        case 3: ta = x2; tb = x2; rsel = 1; break;
        case 4: ta = q1; tb = x2; break;
        default: ta = q2; tb = x1; break;
    }
    const bool self = (pair < 4);
    const float* Ag = ta + (size_t)b * CDIM * N_TOK;
    const float* Bg = tb + (size_t)b * CDIM * N_TOK;

    const int tid  = threadIdx.x;
    const int lane = tid & 31;
    const int wave = tid >> 5;
    const int lo   = lane & 15;
    const int hi   = lane >> 4;

    // 12x12 tile grid of 16x16 tiles; wave w owns rows (w%4)*3..+2, cols (w/4)*6..+5
    const int r0 = (wave & 3) * 3;
    const int c0 = (wave >> 2) * 6;

    v8f acc[18];
    for (int t = 0; t < 18; ++t) acc[t] = zero8();
    float rsum = 0.f;

    const int base = slice * SLICE;
    for (int ch = 0; ch < NCHUNK; ++ch) {
        const int tok0 = base + ch * CH;
        for (int i = tid; i < CDIM * CH; i += 256) {
            int r = i / CH, c = i % CH;
            As[r * LPAD + c] = Ag[(size_t)r * N_TOK + tok0 + c];
            if (!self) Bs[r * LPAD + c] = Bg[(size_t)r * N_TOK + tok0 + c];
        }
        __syncthreads();
        if (self && tid < CDIM) {
            float s = 0.f;
            for (int c = 0; c < CH; ++c) s += As[tid * LPAD + c];
            rsum += s;
        }
        const float* Bsrc = self ? As : Bs;
        for (int kk = 0; kk < CH; kk += 4) {
            const int kc = kk + 2 * hi;
            v2f af[3], bf[6];
            for (int i = 0; i < 3; ++i) {
                const int row = (r0 + i) * 16 + lo;
                af[i].x = As[row * LPAD + kc];
                af[i].y = As[row * LPAD + kc + 1];
            }
            for (int j = 0; j < 6; ++j) {
                const int row = (c0 + j) * 16 + lo;
                bf[j].x = Bsrc[row * LPAD + kc];
                bf[j].y = Bsrc[row * LPAD + kc + 1];
            }
            for (int i = 0; i < 3; ++i)
                for (int j = 0; j < 6; ++j)
                    acc[i * 6 + j] = __builtin_amdgcn_wmma_f32_16x16x4_f32(
                        false, af[i], false, bf[j], (short)0, acc[i * 6 + j],
                        false, false);
        }
        __syncthreads();
    }

    float* Cb = Cacc + ((size_t)b * 6 + pair) * CDIM * CDIM;
    for (int i = 0; i < 3; ++i) {
        for (int j = 0; j < 6; ++j) {
            for (int r = 0; r < 8; ++r) {
                const int row = (r0 + i) * 16 + r + 8 * hi;
                const int col = (c0 + j) * 16 + lo;
                atomicAdd(&Cb[row * CDIM + col], acc[i * 6 + j][r]);
            }
        }
    }
    if (self && tid < CDIM)
        atomicAdd(&Racc[((size_t)b * 4 + rsel) * CDIM + tid], rsum);
}

// ============================================================================
// Kernel 2: U[b][u] = L @ C[b][p]   VALU, tiny
// u: 0:(Wq,C4) 1:(Wq,C5) 2:(Wq,C0) 3:(Wq,C1) 4:(Wk,C2) 5:(Wk,C3)
// ============================================================================
__global__ __launch_bounds__(192) void u_kernel(
    const float* __restrict__ Wq, const float* __restrict__ Wk,
    const float* __restrict__ C, float* __restrict__ U)
{
    const int row = blockIdx.x;          // 0..191
    const int bu  = blockIdx.y;          // 0..47
    const int b = bu / 6, u = bu % 6;
    const float* L = (u <= 3) ? Wq : Wk;
    const int pmap[6] = {4, 5, 0, 1, 2, 3};
    const float* Cp = C + ((size_t)b * 6 + pmap[u]) * CDIM * CDIM;
    const int col = threadIdx.x;
    float s = 0.f;
    for (int k = 0; k < CDIM; ++k)
        s += L[row * CDIM + k] * Cp[k * CDIM + col];
    U[((size_t)b * 6 + u) * CDIM * CDIM + row * CDIM + col] = s;
}

// ============================================================================
// Kernel 3: logits (with bias rank-1 terms), l2-norms, softmax -> A[b*2+dir][h]
// ============================================================================
__global__ __launch_bounds__(256) void attn_kernel(
    const float* __restrict__ Wq, const float* __restrict__ bq,
    const float* __restrict__ Wk, const float* __restrict__ bk,
    const float* __restrict__ scale,
    const float* __restrict__ U, const float* __restrict__ R,
    float* __restrict__ Aout)
{
    const int h = blockIdx.x, dir = blockIdx.y, b = blockIdx.z;
    __shared__ float nq[HD], nk[HD], vq[HD], vk[HD];
    __shared__ float S[HD * HD];
    const float* Ub = U + (size_t)b * 6 * CDIM * CDIM;
    const float* Rb = R + (size_t)b * 4 * CDIM;
    const float* Ucross = Ub + (size_t)(dir == 0 ? 0 : 1) * CDIM * CDIM;
    const float* Uqs    = Ub + (size_t)(dir == 0 ? 2 : 3) * CDIM * CDIM;
    const float* Uks    = Ub + (size_t)(dir == 0 ? 5 : 4) * CDIM * CDIM;
    const float* rq = Rb + (dir == 0 ? 2 : 3) * CDIM;
    const float* rk = Rb + (dir == 0 ? 1 : 0) * CDIM;
    const int tid = threadIdx.x;
    const float Nf = (float)N_TOK;

    if (tid < HD) {
        const int cq = h * HD + tid;
        float v = 0.f, ss = 0.f;
        for (int k = 0; k < CDIM; ++k) {
            v  += Wq[cq * CDIM + k] * rq[k];
            ss += Uqs[cq * CDIM + k] * Wq[cq * CDIM + k];
        }
        ss += 2.f * bq[cq] * v + Nf * bq[cq] * bq[cq];
        vq[tid] = v;
        nq[tid] = fmaxf(sqrtf(fmaxf(ss, 0.f)), 1e-12f);
    } else if (tid < 2 * HD) {
        const int d = tid - HD, ck = h * HD + d;
        float v = 0.f, ss = 0.f;
        for (int k = 0; k < CDIM; ++k) {
            v  += Wk[ck * CDIM + k] * rk[k];
            ss += Uks[ck * CDIM + k] * Wk[ck * CDIM + k];
        }
        ss += 2.f * bk[ck] * v + Nf * bk[ck] * bk[ck];
        vk[d] = v;
        nk[d] = fmaxf(sqrtf(fmaxf(ss, 0.f)), 1e-12f);
    }
    __syncthreads();
    const float sc = scale[h];
    for (int e = tid; e < HD * HD; e += 256) {
        const int c = e / HD, d = e % HD;
        const int cq = h * HD + c, ck = h * HD + d;
        float g = 0.f;
        for (int k = 0; k < CDIM; ++k)
            g += Ucross[cq * CDIM + k] * Wk[ck * CDIM + k];
        g += bq[cq] * vk[d] + bk[ck] * vq[c] + Nf * bq[cq] * bk[ck];
        S[e] = g / (nq[c] * nk[d]) * sc;
    }
    __syncthreads();
    if (tid < HD) {
        float m = -INFINITY;
        for (int d = 0; d < HD; ++d) m = fmaxf(m, S[tid * HD + d]);
        float sum = 0.f;
        for (int d = 0; d < HD; ++d) {
            float e = __expf(S[tid * HD + d] - m);
            S[tid * HD + d] = e; sum += e;
        }
        const float inv = 1.f / sum;
        float* Ar = Aout + (((size_t)(b * 2 + dir) * HEADS + h) * HD + tid) * HD;
        for (int d = 0; d < HD; ++d) Ar[d] = S[tid * HD + d] * inv;
    }
}

// ============================================================================
// Kernel 4: AWv = A_blockdiag @ Wv ; abv = A_blockdiag @ bv
// ============================================================================
__global__ __launch_bounds__(192) void awv_kernel(
    const float* __restrict__ A, const float* __restrict__ Wv,
    const float* __restrict__ bv, float* __restrict__ AWv, float* __restrict__ abv)
{
    const int c = blockIdx.x, bd = blockIdx.y;
    const int h = c / HD, ci = c % HD;
    const float* Ar = A + (((size_t)bd * HEADS + h) * HD + ci) * HD;
    const int j = threadIdx.x;
    float s = 0.f;
    for (int d = 0; d < HD; ++d) s += Ar[d] * Wv[(h * HD + d) * CDIM + j];
    AWv[((size_t)bd * CDIM + c) * CDIM + j] = s;
    if (j == 0) {
        float t = 0.f;
        for (int d = 0; d < HD; ++d) t += Ar[d] * bv[h * HD + d];
        abv[(size_t)bd * CDIM + c] = t;
    }
}

// ============================================================================
// Kernel 5: M = Wproj @ AWv ; cvec = Wproj @ abv
// ============================================================================
__global__ __launch_bounds__(192) void mproj_kernel(
    const float* __restrict__ Wproj, const float* __restrict__ AWv,
    const float* __restrict__ abv, float* __restrict__ Mout, float* __restrict__ cvec)
{
    const int o = blockIdx.x, bd = blockIdx.y;
    const int j = threadIdx.x;
    const float* Ab = AWv + (size_t)bd * CDIM * CDIM;
    float s = 0.f;
    for (int c = 0; c < CDIM; ++c) s += Wproj[o * CDIM + c] * Ab[c * CDIM + j];
    Mout[((size_t)bd * CDIM + o) * CDIM + j] = s;
    if (j == 0) {
        float t = 0.f;
        for (int c = 0; c < CDIM; ++c) t += Wproj[o * CDIM + c] * abv[bd * CDIM + c];
        cvec[bd * CDIM + o] = t;
    }
}

// ============================================================================
// Kernel 6: out[b,dir] = M[b,dir] @ X + cvec   (192x192 @ 192x16384, WMMA f32)
// B-fragment loaded once per k-step from global, reused across 12 row tiles.
// ============================================================================
__global__ __launch_bounds__(256) void outgemm_kernel(
    const float* __restrict__ x1, const float* __restrict__ x2,
    const float* __restrict__ Mmat, const float* __restrict__ cvec,
    float* __restrict__ out)
{
    __shared__ float Ms[CDIM * 193];
    __shared__ float cs[CDIM];
    const int tb = blockIdx.x;   // 0..15 token block (1024 tokens)
    const int bd = blockIdx.y;   // 0..15 : b*2+dir
    const int b = bd >> 1, dir = bd & 1;
    const float* X = (dir == 0 ? x2 : x1) + (size_t)b * CDIM * N_TOK;
    const float* Mg = Mmat + (size_t)bd * CDIM * CDIM;
    float* O = out + (size_t)dir * NB * CDIM * N_TOK + (size_t)b * CDIM * N_TOK;

    const int tid = threadIdx.x;
    const int lane = tid & 31, wave = tid >> 5;
    const int lo = lane & 15, hi = lane >> 4;

    for (int i = tid; i < CDIM * CDIM; i += 256) {
        int r = i / CDIM, c = i % CDIM;
        Ms[r * 193 + c] = Mg[i];
    }
    if (tid < CDIM) cs[tid] = cvec[bd * CDIM + tid];
    __syncthreads();

    for (int j = 0; j < 8; ++j) {
        const int n0 = tb * SLICE + (wave + 8 * j) * 16;
        v8f acc[12];
        for (int t = 0; t < 12; ++t) acc[t] = zero8();
        for (int k0 = 0; k0 < CDIM; k0 += 4) {
            const int kc = k0 + 2 * hi;
            v2f bv;
            bv.x = X[(size_t)kc * N_TOK + n0 + lo];
            bv.y = X[(size_t)(kc + 1) * N_TOK + n0 + lo];
            for (int i0 = 0; i0 < 12; ++i0) {
                v2f av;
                av.x = Ms[(i0 * 16 + lo) * 193 + kc];
                av.y = Ms[(i0 * 16 + lo) * 193 + kc + 1];
                acc[i0] = __builtin_amdgcn_wmma_f32_16x16x4_f32(
                    false, av, false, bv, (short)0, acc[i0], false, false);
            }
        }
        for (int i0 = 0; i0 < 12; ++i0) {
            for (int r = 0; r < 8; ++r) {
                const int row = i0 * 16 + r + 8 * hi;
                O[(size_t)row * N_TOK + n0 + lo] = acc[i0][r] + cs[row];
            }
        }
    }
}

// ============================================================================
extern "C" void kernel_launch(void* const* d_in, const int* in_sizes, int n_in,
                              void* d_out, int out_size, void* d_ws, size_t ws_size,
                              hipStream_t stream) {
    (void)in_sizes; (void)n_in; (void)out_size; (void)ws_size;
    const float* x1    = (const float*)d_in[0];
    const float* x2    = (const float*)d_in[1];
    const float* q1    = (const float*)d_in[2];
    const float* q2    = (const float*)d_in[3];
    const float* Wq    = (const float*)d_in[4];
    const float* bq    = (const float*)d_in[5];
    const float* Wk    = (const float*)d_in[6];
    const float* bk    = (const float*)d_in[7];
    const float* Wv    = (const float*)d_in[8];
    const float* bv    = (const float*)d_in[9];
    const float* scale = (const float*)d_in[10];
    const float* Wproj = (const float*)d_in[11];

    float* ws  = (float*)d_ws;
    float* C   = ws + OFF_C;
    float* R   = ws + OFF_R;
    float* U   = ws + OFF_U;
    float* A   = ws + OFF_A;
    float* AWv = ws + OFF_AWV;
    float* abv = ws + OFF_ABV;
    float* Mm  = ws + OFF_M;
    float* cv  = ws + OFF_CV;

    // zero the atomic accumulators (C + R regions)
    hipMemsetAsync(ws, 0, OFF_U * sizeof(float), stream);

    gram_kernel   <<<dim3(16, 6, 8), 256, 0, stream>>>(x1, x2, q1, q2, C, R);
    u_kernel      <<<dim3(192, 48),  192, 0, stream>>>(Wq, Wk, C, U);
    attn_kernel   <<<dim3(8, 2, 8),  256, 0, stream>>>(Wq, bq, Wk, bk, scale, U, R, A);
    awv_kernel    <<<dim3(192, 16),  192, 0, stream>>>(A, Wv, bv, AWv, abv);
    mproj_kernel  <<<dim3(192, 16),  192, 0, stream>>>(Wproj, AWv, abv, Mm, cv);
    outgemm_kernel<<<dim3(16, 16),   256, 0, stream>>>(x1, x2, Mm, cv, (float*)d_out);
}